// TTM_5703716569220
// MI455X (gfx1250) — compile-verified
//
#include <hip/hip_runtime.h>
#include <hip/hip_bf16.h>
#include <math.h>
#include <stdint.h>

typedef __attribute__((ext_vector_type(2))) float v2f;
typedef __attribute__((ext_vector_type(8))) float v8f;

// ---- problem constants (match reference) ----
#define NB   32      // B
#define NA   32      // A
#define NC   16      // C
#define NT   2000    // T
#define AVGF 10
#define NK   3
#define NH   8
#define NM   512     // M = C*A
#define NDH  64
#define NS   11      // AVGF+1
#define SP   16      // padded sequence rows

__device__ __forceinline__ v8f wmma4(v2f a, v2f b, v8f c) {
  // D = A(16x4 f32) * B(4x16 f32) + C(16x16 f32)
  return __builtin_amdgcn_wmma_f32_16x16x4_f32(false, a, false, b, (short)0, c,
                                               false, false);
}

// CDNA5 async global->LDS copy (tracked on ASYNCcnt), 16B per lane.
__device__ __forceinline__ void async_copy_b128(uint32_t lds_byte_addr,
                                                const void* gaddr) {
  asm volatile("global_load_async_to_lds_b128 %0, %1, off"
               :
               : "v"(lds_byte_addr), "v"(gaddr)
               : "memory");
}
__device__ __forceinline__ void wait_async0() {
  asm volatile("s_wait_asynccnt 0" ::: "memory");
}
// generic pointer into __shared__ -> wave-relative LDS byte offset
__device__ __forceinline__ uint32_t lds_off32(const void* p) {
  return (uint32_t)(uintptr_t)p;
}

// ---------------- segment mean: x[B,A,C,T] -> altx[B,16(pad),512] ----------------
__global__ __launch_bounds__(64)
void seg_mean_kernel(const float* __restrict__ x, float* __restrict__ altx) {
  __shared__ float red[64];
  int id = blockIdx.x;
  int i = id % AVGF;
  int c = (id / AVGF) % NC;
  int a = (id / (AVGF * NC)) % NA;
  int b = id / (AVGF * NC * NA);
  const float* xp = x + (((size_t)b * NA + a) * NC + c) * NT + (size_t)i * 200;
  float s = 0.f;
  for (int t = threadIdx.x; t < 200; t += 64) s += xp[t];
  red[threadIdx.x] = s;
  __syncthreads();
  for (int st = 32; st > 0; st >>= 1) {
    if (threadIdx.x < st) red[threadIdx.x] += red[threadIdx.x + st];
    __syncthreads();
  }
  if (threadIdx.x == 0)
    altx[((size_t)b * SP + i) * NM + c * NA + a] = red[0] * (1.0f / 200.0f);
}

__global__ void fill0_kernel(float* __restrict__ p, int n) {
  int i = blockIdx.x * blockDim.x + threadIdx.x;
  if (i < n) p[i] = 0.f;
}

// ---------------- generic fp32 WMMA GEMM ----------------
// out[b, row, n] = sum_m act[b,row,m] * W[b?, n, m]  (+bias[n]) (gelu) (+resid)
// act: [B, 16, Kd] (pad rows zero);  W rows = output n, cols = reduction m.
// grid = (B, Nd/128); block = 256 (8 waves, one 16-col tile per wave)
__global__ __launch_bounds__(256)
void wmma_gemm_f32(const float* __restrict__ act, const float* __restrict__ W,
                   const float* __restrict__ bias,
                   const float* __restrict__ resid, float* __restrict__ out,
                   int Kd, int Nd, long wStrideB, int gelu) {
  __shared__ float lds[SP * (512 + 4)];
  const int b    = blockIdx.x;
  const int wave = threadIdx.x >> 5;
  const int lane = threadIdx.x & 31;
  const int lr   = lane & 15;
  const int hi   = lane >> 4;  // 0 or 1
  const int n0   = (blockIdx.y * 8 + wave) * 16;

  const float* actB = act + (size_t)b * SP * Kd;
  const float* Wr   = W + (size_t)b * wStrideB + (size_t)(n0 + lr) * Kd + 2 * hi;

  v8f acc = {};
  for (int kc = 0; kc < Kd; kc += 512) {
    // async-stage the 16x512 activation chunk into LDS (ASYNCcnt path)
#pragma unroll
    for (int it = 0; it < 8; ++it) {
      int fidx = (threadIdx.x + it * 256) * 4;  // float index, 16B granules
      int r = fidx >> 9;
      int cc = fidx & 511;
      async_copy_b128(lds_off32(&lds[r * 516 + cc]),
                      actB + (size_t)r * Kd + kc + cc);
    }
    wait_async0();
    __syncthreads();

    const float* Ar = &lds[lr * 516 + 2 * hi];
    const float* Wk = Wr + kc;
#pragma unroll 4
    for (int k0 = 0; k0 < 512; k0 += 4) {
      if ((k0 & 127) == 0)  // stream-prefetch the weight rows one chunk ahead
        __builtin_prefetch(Wk + k0 + 512, 0, 1);
      float2 av = *reinterpret_cast<const float2*>(Ar + k0);
      float2 bv = *reinterpret_cast<const float2*>(Wk + k0);
      v2f a;  a[0] = av.x;  a[1] = av.y;
      v2f bb; bb[0] = bv.x; bb[1] = bv.y;
      acc = wmma4(a, bb, acc);
    }
    __syncthreads();
  }
  const int n = n0 + lr;
  const float bn = bias ? bias[n] : 0.0f;
#pragma unroll
  for (int r = 0; r < 8; ++r) {
    int row = r + 8 * hi;  // D layout: vgpr r holds rows r (lanes 0-15) / r+8 (16-31)
    float v = acc[r] + bn;
    if (gelu) v = 0.5f * v * (1.0f + erff(v * 0.70710678118654752f));
    size_t o = ((size_t)b * SP + row) * Nd + n;
    if (resid) v += resid[o];
    out[o] = v;
  }
}

// ---------------- attention (no softmax), one wave per (b,h) ----------------
// qkv: [B,16,1536] cols = x*512 + h*64 + d ; imv out: [B,16,512] cols = h*64+d
__global__ __launch_bounds__(32)
void attn_kernel(const float* __restrict__ qkv, float* __restrict__ imv) {
  __shared__ float ats[16 * 17];
  const int h = blockIdx.x & (NH - 1);
  const int b = blockIdx.x / NH;
  const int lane = threadIdx.x;
  const int lr = lane & 15;
  const int hi = lane >> 4;
  const float scale = 0.125f;  // 1/sqrt(64)

  const float* qrow = qkv + ((size_t)b * SP + lr) * (3 * NM) + h * NDH + 2 * hi;
  const float* krow = qrow + NM;
  v8f at = {};
#pragma unroll
  for (int k0 = 0; k0 < NDH; k0 += 4) {
    float2 qv = *reinterpret_cast<const float2*>(qrow + k0);
    float2 kv = *reinterpret_cast<const float2*>(krow + k0);
    v2f a;  a[0] = qv.x * scale; a[1] = qv.y * scale;
    v2f bb; bb[0] = kv.x;        bb[1] = kv.y;
    at = wmma4(a, bb, at);
  }
#pragma unroll
  for (int r = 0; r < 8; ++r) ats[(r + 8 * hi) * 17 + lr] = at[r];
  __syncthreads();

  const float* vbase = qkv + (size_t)b * SP * (3 * NM) + 2 * NM + h * NDH;
#pragma unroll
  for (int d0 = 0; d0 < NDH; d0 += 16) {
    v8f o = {};
#pragma unroll
    for (int k0 = 0; k0 < 16; k0 += 4) {
      int j0 = k0 + 2 * hi;
      v2f a;  a[0] = ats[lr * 17 + j0]; a[1] = ats[lr * 17 + j0 + 1];
      v2f bb;
      bb[0] = vbase[(size_t)j0 * (3 * NM) + d0 + lr];
      bb[1] = vbase[(size_t)(j0 + 1) * (3 * NM) + d0 + lr];
      o = wmma4(a, bb, o);
    }
#pragma unroll
    for (int r = 0; r < 8; ++r)
      imv[((size_t)b * SP + r + 8 * hi) * NM + h * NDH + d0 + lr] = o[r];
  }
}

// ---------------- layernorm over last dim (512), rows 0..10 ----------------
// grid = B*16, block = 256. outRows==16: pad rows written as zero (padded buf);
// outRows==11: pad rows skipped (final output).
__global__ __launch_bounds__(256)
void ln_kernel(const float* __restrict__ in, const float* __restrict__ gw,
               const float* __restrict__ gb, float* __restrict__ out,
               int outRows) {
  __shared__ float red[256];
  __shared__ float stat[2];
  const int row = blockIdx.x & 15;
  const int b = blockIdx.x >> 4;
  if (row >= NS) {
    if (outRows == SP) {
      float* op = out + ((size_t)b * SP + row) * NM;
      op[threadIdx.x] = 0.f;
      op[threadIdx.x + 256] = 0.f;
    }
    return;
  }
  const float* ip = in + ((size_t)b * SP + row) * NM;
  float v0 = ip[threadIdx.x], v1 = ip[threadIdx.x + 256];
  red[threadIdx.x] = v0 + v1;
  __syncthreads();
  for (int st = 128; st > 0; st >>= 1) {
    if (threadIdx.x < st) red[threadIdx.x] += red[threadIdx.x + st];
    __syncthreads();
  }
  if (threadIdx.x == 0) stat[0] = red[0] * (1.0f / 512.0f);
  __syncthreads();
  const float mu = stat[0];
  float d0 = v0 - mu, d1 = v1 - mu;
  red[threadIdx.x] = d0 * d0 + d1 * d1;
  __syncthreads();
  for (int st = 128; st > 0; st >>= 1) {
    if (threadIdx.x < st) red[threadIdx.x] += red[threadIdx.x + st];
    __syncthreads();
  }
  if (threadIdx.x == 0) stat[1] = rsqrtf(red[0] * (1.0f / 512.0f) + 1e-5f);
  __syncthreads();
  const float rs = stat[1];
  float* op = out + ((size_t)b * outRows + row) * NM;
  op[threadIdx.x]       = d0 * rs * gw[threadIdx.x]       + gb[threadIdx.x];
  op[threadIdx.x + 256] = d1 * rs * gw[threadIdx.x + 256] + gb[threadIdx.x + 256];
}

// ---------------- sav = concat(cls, weight@altx) + bias; pad rows zero ----------
__global__ __launch_bounds__(256)
void sav_assemble_kernel(const float* __restrict__ tmp0,
                         const float* __restrict__ cls,
                         const float* __restrict__ bias,
                         float* __restrict__ sav) {
  const int b = blockIdx.x;
  for (int idx = threadIdx.x; idx < SP * NM; idx += 256) {
    int row = idx >> 9, m = idx & 511;
    float v;
    if (row == 0)
      v = cls[(size_t)b * NM + m] + bias[(size_t)b * NS * NM + m];
    else if (row <= AVGF)
      v = tmp0[((size_t)b * SP + row - 1) * NM + m] +
          bias[((size_t)b * NS + row) * NM + m];
    else
      v = 0.f;
    sav[((size_t)b * SP + row) * NM + m] = v;
  }
}

extern "C" void kernel_launch(void* const* d_in, const int* in_sizes, int n_in,
                              void* d_out, int out_size, void* d_ws,
                              size_t ws_size, hipStream_t stream) {
  const float* x      = (const float*)d_in[0];
  const float* weight = (const float*)d_in[1];
  const float* bias   = (const float*)d_in[2];
  const float* cls    = (const float*)d_in[3];
  const float* Wqkv   = (const float*)d_in[4];
  const float* Wo     = (const float*)d_in[5];
  const float* ln1w   = (const float*)d_in[6];
  const float* ln1b   = (const float*)d_in[7];
  const float* ln2w   = (const float*)d_in[8];
  const float* ln2b   = (const float*)d_in[9];
  const float* fc1w   = (const float*)d_in[10];
  const float* fc1b   = (const float*)d_in[11];
  const float* fc2w   = (const float*)d_in[12];
  const float* fc2b   = (const float*)d_in[13];
  const float* lnxw   = (const float*)d_in[14];
  const float* lnxb   = (const float*)d_in[15];
  float* out = (float*)d_out;

  // workspace layout (floats)
  float* ws    = (float*)d_ws;
  float* altx  = ws;                           // 32*16*512
  float* tmp0  = altx  + NB * SP * NM;         // 32*16*512
  float* sav   = tmp0  + NB * SP * NM;         // 32*16*512
  float* lnbuf = sav   + NB * SP * NM;         // 32*16*512
  float* qkvb  = lnbuf + NB * SP * NM;         // 32*16*1536
  float* imvb  = qkvb  + NB * SP * 3 * NM;     // 32*16*512
  float* h1    = imvb  + NB * SP * NM;         // 32*16*2048

  const int altxN = NB * SP * NM;
  fill0_kernel<<<(altxN + 255) / 256, 256, 0, stream>>>(altx, altxN);
  seg_mean_kernel<<<NB * NA * NC * AVGF, 64, 0, stream>>>(x, altx);

  // sav0 = einsum('blm,bim->bil'): per-batch W (stride M*M)
  wmma_gemm_f32<<<dim3(NB, 4), 256, 0, stream>>>(
      altx, weight, nullptr, nullptr, tmp0, NM, NM, (long)NM * NM, 0);
  sav_assemble_kernel<<<NB, 256, 0, stream>>>(tmp0, cls, bias, sav);

  for (int k = 0; k < NK; ++k) {
    ln_kernel<<<NB * SP, 256, 0, stream>>>(sav, ln1w + k * NM, ln1b + k * NM,
                                           lnbuf, SP);
    wmma_gemm_f32<<<dim3(NB, 12), 256, 0, stream>>>(
        lnbuf, Wqkv + (size_t)k * 3 * NM * NM, nullptr, nullptr, qkvb, NM,
        3 * NM, 0, 0);
    attn_kernel<<<NB * NH, 32, 0, stream>>>(qkvb, imvb);
    wmma_gemm_f32<<<dim3(NB, 4), 256, 0, stream>>>(
        imvb, Wo + (size_t)k * NM * NM, nullptr, sav, sav, NM, NM, 0, 0);
    ln_kernel<<<NB * SP, 256, 0, stream>>>(sav, ln2w + k * NM, ln2b + k * NM,
                                           lnbuf, SP);
    wmma_gemm_f32<<<dim3(NB, 16), 256, 0, stream>>>(
        lnbuf, fc1w + (size_t)k * 4 * NM * NM, fc1b + (size_t)k * 4 * NM,
        nullptr, h1, NM, 4 * NM, 0, 1);
    wmma_gemm_f32<<<dim3(NB, 4), 256, 0, stream>>>(
        h1, fc2w + (size_t)k * NM * 4 * NM, fc2b + (size_t)k * NM, sav, sav,
        4 * NM, NM, 0, 0);
  }
  ln_kernel<<<NB * SP, 256, 0, stream>>>(sav, lnxw, lnxb, out, NS);
}